// MeshLoss_42623255445862
// MI455X (gfx1250) — compile-verified
//
#include <hip/hip_runtime.h>

#define B_SZ 4
#define V_SZ 6890
#define F_SZ 13776
#define N_SZ 4096

typedef __attribute__((ext_vector_type(2))) float v2f;
typedef __attribute__((ext_vector_type(8))) float v8f;

// ---------------- RNG: counter-based hash, same stream for pred & gt (mirrors key reuse) ----
__device__ __forceinline__ float rng01(unsigned b, unsigned i, unsigned s) {
    unsigned long long x =
        (((unsigned long long)(b * 0x9E3779B1u + s * 0x85EBCA77u + 0x2Au)) << 32) |
        (unsigned long long)(i * 0xC2B2AE3Du + 0x27D4EB2Fu);
    x ^= x >> 30; x *= 0xBF58476D1CE4E5B9ull;
    x ^= x >> 27; x *= 0x94D049BB133111EBull;
    x ^= x >> 31;
    return (float)(unsigned)((x >> 40) & 0xFFFFFFu) * (1.0f / 16777216.0f);
}

// ---------------- 1) face areas ----------------
__global__ void area_kernel(const float* __restrict__ verts, const int* __restrict__ faces,
                            float* __restrict__ area) {
    int gid = blockIdx.x * blockDim.x + threadIdx.x;
    if (gid >= B_SZ * F_SZ) return;
    int b = gid / F_SZ, f = gid % F_SZ;
    const float* vb = verts + (size_t)b * V_SZ * 3;
    const int*   fb = faces + (size_t)b * F_SZ * 3 + (size_t)f * 3;
    int i0 = fb[0], i1 = fb[1], i2 = fb[2];
    float ax = vb[i0*3+0], ay = vb[i0*3+1], az = vb[i0*3+2];
    float e1x = vb[i1*3+0]-ax, e1y = vb[i1*3+1]-ay, e1z = vb[i1*3+2]-az;
    float e2x = vb[i2*3+0]-ax, e2y = vb[i2*3+1]-ay, e2z = vb[i2*3+2]-az;
    float cx = e1y*e2z - e1z*e2y;
    float cy = e1z*e2x - e1x*e2z;
    float cz = e1x*e2y - e1y*e2x;
    area[gid] = 0.5f * sqrtf(cx*cx + cy*cy + cz*cz);
}

// ---------------- 2) in-place inclusive scan per batch (one block/batch) ----------------
__global__ void scan_kernel(float* __restrict__ area) {
    __shared__ float s[256];
    __shared__ float carry;
    float* a = area + (size_t)blockIdx.x * F_SZ;
    int t = threadIdx.x;
    if (t == 0) carry = 0.0f;
    __syncthreads();
    for (int base = 0; base < F_SZ; base += 256) {
        int idx = base + t;
        float x = (idx < F_SZ) ? a[idx] : 0.0f;
        s[t] = x; __syncthreads();
        for (int off = 1; off < 256; off <<= 1) {
            float u = (t >= off) ? s[t - off] : 0.0f;
            __syncthreads();
            s[t] += u; __syncthreads();
        }
        float incl = s[t] + carry;
        if (idx < F_SZ) a[idx] = incl;
        __syncthreads();
        if (t == 0) carry += s[255];
        __syncthreads();
    }
}

// ---------------- 3) sample points; store float4{x,y,z,|p|^2} ----------------
__global__ void sample_kernel(const float* __restrict__ verts, const int* __restrict__ faces,
                              const float* __restrict__ cdf, float4* __restrict__ pc) {
    int gid = blockIdx.x * blockDim.x + threadIdx.x;
    if (gid >= B_SZ * N_SZ) return;
    int b = gid / N_SZ, i = gid % N_SZ;
    const float* cb = cdf + (size_t)b * F_SZ;
    float target = rng01(b, i, 0) * cb[F_SZ - 1];
    int lo = 0, hi = F_SZ - 1;
    while (lo < hi) { int mid = (lo + hi) >> 1; if (cb[mid] < target) lo = mid + 1; else hi = mid; }
    const float* vb = verts + (size_t)b * V_SZ * 3;
    const int*   fb = faces + (size_t)b * F_SZ * 3 + (size_t)lo * 3;
    int i0 = fb[0], i1 = fb[1], i2 = fb[2];
    float r = sqrtf(rng01(b, i, 1));
    float w = rng01(b, i, 2);
    float a0 = 1.0f - r, a1 = r * (1.0f - w), a2 = r * w;
    float px = a0*vb[i0*3+0] + a1*vb[i1*3+0] + a2*vb[i2*3+0];
    float py = a0*vb[i0*3+1] + a1*vb[i1*3+1] + a2*vb[i2*3+1];
    float pz = a0*vb[i0*3+2] + a1*vb[i1*3+2] + a2*vb[i2*3+2];
    pc[gid] = make_float4(px, py, pz, px*px + py*py + pz*pz);
}

// ---------------- 4) chamfer row-mins via V_WMMA_F32_16X16X4_F32 ----------------
// Spare K slot carries the row norm:
//   A[i,:] = (-2x_i, -2y_i, -2z_i, |p_i|^2)   B[:,j] = (qx_j, qy_j, qz_j, 1)
// => D[i,j] = |p_i|^2 - 2 p_i.q_j ; full distance = D[i,j] + |q_j|^2 (|q|^2 uniform per lane).
// A layout (16x4 f32): lanes 0-15 hold {K0,K1}; lanes 16-31 hold {K2,K3}, row = lane&15.
// B layout (4x16 f32): lanes 0-15 hold {K0,K1}; lanes 16-31 hold {K2,K3}, col = lane&15.
// C/D layout: VGPR r, lanes 0-15 -> (M=r, N=lane); lanes 16-31 -> (M=r+8, N=lane-16).
__global__ void chamfer_kernel(const float4* __restrict__ P, const float4* __restrict__ Q,
                               float* __restrict__ rowmin) {
    const int lane = threadIdx.x & 31;       // full wave, EXEC all ones, uniform flow
    const int wid  = threadIdx.x >> 5;       // 4 waves/block, one 16-row tile each
    const int b = blockIdx.y;
    const int rowBase = blockIdx.x * 64 + wid * 16;
    const bool hi = lane >= 16;
    const float4* Pb = P + (size_t)b * N_SZ;
    const float4* Qb = Q + (size_t)b * N_SZ;

    float4 p = Pb[rowBase + (lane & 15)];
    v2f a;
    a.x = hi ? (-2.0f * p.z) : (-2.0f * p.x);
    a.y = hi ? p.w           : (-2.0f * p.y);

    float minv[8];
#pragma unroll
    for (int r = 0; r < 8; ++r) minv[r] = 3.402823466e+38f;

    const v8f czero = {};
#pragma unroll 4
    for (int colBase = 0; colBase < N_SZ; colBase += 16) {
        float4 q = Qb[colBase + (lane & 15)];
        v2f bb;
        bb.x = hi ? q.z  : q.x;
        bb.y = hi ? 1.0f : q.y;
        v8f dmat = __builtin_amdgcn_wmma_f32_16x16x4_f32(
            /*neg_a=*/false, a, /*neg_b=*/false, bb,
            /*c_mod=*/(short)0, czero, /*reuse_a=*/false, /*reuse_b=*/false);
#pragma unroll
        for (int r = 0; r < 8; ++r)
            minv[r] = fminf(minv[r], dmat[r] + q.w);
    }

    // min across the 16 lanes of each half-wave (same D row)
#pragma unroll
    for (int r = 0; r < 8; ++r) {
        float m = minv[r];
#pragma unroll
        for (int off = 8; off > 0; off >>= 1)
            m = fminf(m, __shfl_xor(m, off, 16));
        if ((lane & 15) == 0)
            rowmin[(size_t)b * N_SZ + rowBase + r + (hi ? 8 : 0)] = m;
    }
}

// ---------------- 5) final mean reduction ----------------
__global__ void reduce_kernel(const float* __restrict__ vals, int n, float scale,
                              float* __restrict__ out) {
    __shared__ float s[256];
    float acc = 0.0f;
    for (int i = threadIdx.x; i < n; i += 256) acc += vals[i];
    s[threadIdx.x] = acc; __syncthreads();
    for (int off = 128; off > 0; off >>= 1) {
        if (threadIdx.x < off) s[threadIdx.x] += s[threadIdx.x + off];
        __syncthreads();
    }
    if (threadIdx.x == 0) out[0] = s[0] * scale;
}

extern "C" void kernel_launch(void* const* d_in, const int* in_sizes, int n_in,
                              void* d_out, int out_size, void* d_ws, size_t ws_size,
                              hipStream_t stream) {
    const float* pred_v = (const float*)d_in[0];
    const int*   pred_f = (const int*)d_in[1];
    const float* gt_v   = (const float*)d_in[2];
    const int*   gt_f   = (const int*)d_in[3];
    float* out = (float*)d_out;
    (void)in_sizes; (void)n_in; (void)out_size; (void)ws_size;

    // workspace layout (floats): cdfP[B*F] cdfG[B*F] pcP[B*N*4] pcG[B*N*4] d1[B*N] d2[B*N]
    float* cdfP = (float*)d_ws;
    float* cdfG = cdfP + (size_t)B_SZ * F_SZ;
    float4* pcP = (float4*)(cdfG + (size_t)B_SZ * F_SZ);
    float4* pcG = pcP + (size_t)B_SZ * N_SZ;
    float* d1   = (float*)(pcG + (size_t)B_SZ * N_SZ);
    // d2 contiguous after d1 so one reduction covers both

    {
        int n = B_SZ * F_SZ;
        area_kernel<<<(n + 255) / 256, 256, 0, stream>>>(pred_v, pred_f, cdfP);
        area_kernel<<<(n + 255) / 256, 256, 0, stream>>>(gt_v,   gt_f,   cdfG);
    }
    scan_kernel<<<B_SZ, 256, 0, stream>>>(cdfP);
    scan_kernel<<<B_SZ, 256, 0, stream>>>(cdfG);

    {
        int n = B_SZ * N_SZ;
        sample_kernel<<<(n + 255) / 256, 256, 0, stream>>>(pred_v, pred_f, cdfP, pcP);
        sample_kernel<<<(n + 255) / 256, 256, 0, stream>>>(gt_v,   gt_f,   cdfG, pcG);
    }

    {
        dim3 grid(N_SZ / 64, B_SZ);   // 4 waves/block, one 16-row tile per wave
        chamfer_kernel<<<grid, 128, 0, stream>>>(pcP, pcG, d1);
        chamfer_kernel<<<grid, 128, 0, stream>>>(pcG, pcP, d1 + (size_t)B_SZ * N_SZ);
    }

    // loss = CHAMFER_W * (mean(d1) + mean(d2)) = (sum d1 + sum d2) / (B*N)
    reduce_kernel<<<1, 256, 0, stream>>>(d1, 2 * B_SZ * N_SZ, 1.0f / (float)(B_SZ * N_SZ), out);
}